// Embedding_65712999629186
// MI455X (gfx1250) — compile-verified
//
#include <hip/hip_runtime.h>

typedef __attribute__((ext_vector_type(2))) float v2f;
typedef __attribute__((ext_vector_type(8))) float v8f;

#define FIELD_DIMS 100000
#define FACTORS 16
#define BATCH 1024
#define MTILES (BATCH / 16)   // 64

// ---------------------------------------------------------------------------
// Kernel 1: one wave32 per block. grid.x = MTILES * nchunks.
// Each wave accumulates two 16x16 f32 tiles over its K-chunk via
// V_WMMA_F32_16X16X4_F32:  S += A*B  (A = mask frag, B = W slab)
//                          T += A*B2 (B2 = W*W slab)
// and writes both partial tiles to workspace (deterministic, no atomics).
// ---------------------------------------------------------------------------
__global__ __launch_bounds__(32)
void fm_partial_kernel(const int* __restrict__ x0, const float* __restrict__ W,
                       float* __restrict__ part, int chunk_fields, int nchunks)
{
    const int blk   = blockIdx.x;
    const int mtile = blk / nchunks;
    const int chunk = blk % nchunks;
    const int lane  = threadIdx.x;     // 0..31
    const int half  = lane >> 4;       // 0 or 1
    const int mrow  = lane & 15;       // row within 16-row tile / B column

    int f_begin = chunk * chunk_fields;
    int f_end   = f_begin + chunk_fields;
    if (f_end > FIELD_DIMS) f_end = FIELD_DIMS;
    const int steps = (f_end - f_begin) >> 2;   // 4 fields (K) per WMMA step

    // A-fragment stream: lane L holds mask[row, f0 + 2*half + {0,1}]
    const int row = mtile * 16 + mrow;
    const int* xp = x0 + (long long)row * FIELD_DIMS + f_begin + 2 * half;

    // B-fragment stream: lane L holds W[f0 + 2*half + {0,1}, mrow]
    const float* wp = W + (long long)(f_begin + 2 * half) * FACTORS + mrow;

    v8f accS = {0.f, 0.f, 0.f, 0.f, 0.f, 0.f, 0.f, 0.f};
    v8f accT = {0.f, 0.f, 0.f, 0.f, 0.f, 0.f, 0.f, 0.f};

    for (int s = 0; s < steps; ++s) {
        int2 xv = *(const int2*)xp;                // 8B/lane; wave = 256B of x0
        v2f a;
        a.x = (xv.x > 0) ? 1.0f : 0.0f;
        a.y = (xv.y > 0) ? 1.0f : 0.0f;

        float b0 = wp[0];
        float b1 = wp[FACTORS];
        v2f b  = {b0, b1};
        v2f b2 = {b0 * b0, b1 * b1};

        // 8 args: (neg_a, A, neg_b, B, c_mod, C, reuse_a, reuse_b)
        accS = __builtin_amdgcn_wmma_f32_16x16x4_f32(
            false, a, false, b,  (short)0, accS, false, false);
        accT = __builtin_amdgcn_wmma_f32_16x16x4_f32(
            false, a, false, b2, (short)0, accT, false, false);

        xp += 4;
        wp += 4 * FACTORS;
    }

    // Partial tile layout: [mtile][chunk][S|T][vreg*32 + lane]
    float* outp = part + (long long)(mtile * nchunks + chunk) * 2 * 256;
#pragma unroll
    for (int i = 0; i < 8; ++i) {
        outp[i * 32 + lane]       = accS[i];   // coalesced 128B stores
        outp[256 + i * 32 + lane] = accT[i];
    }
}

// ---------------------------------------------------------------------------
// Kernel 2: 64 blocks x 256 threads. Thread (m,n) sums its (m,n) element of
// S and T over all chunks (fixed order -> deterministic), then LDS row-reduce:
// out[b] = 0.5 * (sum_n S[b,n]^2 - sum_n T[b,n])
// ---------------------------------------------------------------------------
__global__ __launch_bounds__(256)
void fm_reduce_kernel(const float* __restrict__ part, float* __restrict__ out,
                      int nchunks)
{
    __shared__ float red_s[256];
    __shared__ float red_t[256];

    const int mtile = blockIdx.x;
    const int t = threadIdx.x;
    const int m = t >> 4;
    const int n = t & 15;
    // WMMA f32 C/D layout: element (m,n) lives at vgpr=(m&7), lane=n+16*(m>>3)
    const int flat = (m & 7) * 32 + n + 16 * (m >> 3);

    const float* base = part + (long long)mtile * nchunks * 2 * 256;
    float s = 0.f, q = 0.f;
    for (int c = 0; c < nchunks; ++c) {
        s += base[(long long)c * 512 + flat];
        q += base[(long long)c * 512 + 256 + flat];
    }
    red_s[m * 16 + n] = s * s;
    red_t[m * 16 + n] = q;
    __syncthreads();

    if (t < 16) {
        float ss = 0.f, qq = 0.f;
#pragma unroll
        for (int k = 0; k < 16; ++k) {
            ss += red_s[t * 16 + k];
            qq += red_t[t * 16 + k];
        }
        out[mtile * 16 + t] = 0.5f * (ss - qq);
    }
}

// ---------------------------------------------------------------------------
extern "C" void kernel_launch(void* const* d_in, const int* in_sizes, int n_in,
                              void* d_out, int out_size, void* d_ws, size_t ws_size,
                              hipStream_t stream)
{
    (void)in_sizes; (void)n_in; (void)out_size;
    const int*   x0 = (const int*)d_in[0];    // [1024, 100000] int32 in {0,1}
    const float* W  = (const float*)d_in[1];  // [100000, 16] f32
    float* out  = (float*)d_out;              // [1024] f32
    float* part = (float*)d_ws;

    // Choose K-split deterministically from ws_size. Each chunk needs
    // MTILES * 2 tiles * 256 floats = 128 KiB of workspace.
    const size_t per_chunk = (size_t)MTILES * 2 * 256 * sizeof(float);
    int nchunks = (int)(ws_size / per_chunk);
    if (nchunks > 100) nchunks = 100;   // 6400 waves: plenty of occupancy
    if (nchunks < 1)   nchunks = 1;

    // chunk size: multiple of 4 fields (K granularity), covers FIELD_DIMS
    int chunk_fields = ((FIELD_DIMS + nchunks - 1) / nchunks + 3) & ~3;
    nchunks = (FIELD_DIMS + chunk_fields - 1) / chunk_fields;

    dim3 g1(MTILES * nchunks), b1(32);
    fm_partial_kernel<<<g1, b1, 0, stream>>>(x0, W, part, chunk_fields, nchunks);

    dim3 g2(MTILES), b2(256);
    fm_reduce_kernel<<<g2, b2, 0, stream>>>(part, out, nchunks);
}